// MinCutPool_56573309223697
// MI455X (gfx1250) — compile-verified
//
#include <hip/hip_runtime.h>
#include <hip/hip_bf16.h>

typedef float v2f __attribute__((ext_vector_type(2)));
typedef float v8f __attribute__((ext_vector_type(8)));

#define N_NODES 8192
#define D_FEAT  256
#define K_CL    64
#define E_EDGES 262144
#define NPAIR   (K_CL * (K_CL - 1))   // 4032
#define EPS_VAL 1e-8f

// ---- d_out layout (floats, concatenated in reference return order) ----
#define OFF_PX 0                          // pooled_x: 64*256
#define OFF_EI (K_CL * D_FEAT)            // pooled_edge_index: 2*4032 (as float)
#define OFF_EW (OFF_EI + 2 * NPAIR)       // pooled_edge_weight: 4032
#define OFF_S  (OFF_EW + NPAIR)           // S: 8192*64

// ---- workspace layout (floats) ----
#define WS_XT 0                                   // x_t: 8192*256 (8 MB)
#define WS_T  (N_NODES * D_FEAT)                  // T = (B+B^T)S: 8192*64 (2 MB)
#define WS_M  (WS_T + N_NODES * K_CL)             // Mraw = S^T T: 64*64
#define WS_CS (WS_M + K_CL * K_CL)                // colsum(S): 64

// fp32 WMMA: D(16x16,f32) = A(16x4,f32) * B(4x16,f32) + C
__device__ inline v8f wmma_f32(v2f a, v2f b, v8f c) {
    return __builtin_amdgcn_wmma_f32_16x16x4_f32(
        /*neg_a=*/false, a, /*neg_b=*/false, b,
        /*c_mod=*/(short)0, c, /*reuse_a=*/false, /*reuse_b=*/false);
}

// -------- init: zero the atomic accumulators (harness poisons buffers) -----
__global__ void k_zero(float* __restrict__ T, float* __restrict__ Mraw,
                       float* __restrict__ colsum, float* __restrict__ px) {
    int idx = blockIdx.x * blockDim.x + threadIdx.x;
    if (idx < N_NODES * K_CL) T[idx] = 0.f;
    if (idx < K_CL * K_CL)    Mraw[idx] = 0.f;
    if (idx < K_CL)           colsum[idx] = 0.f;
    if (idx < K_CL * D_FEAT)  px[idx] = 0.f;
}

// -------- S = softmax(x @ Wa + ba); also colsum(S) via atomics ------------
// 128 threads = 4 waves; block handles 16 rows x 64 cols; wave w -> col tile.
__global__ void k_assign_softmax(const float* __restrict__ x,
                                 const float* __restrict__ Wa,
                                 const float* __restrict__ ba,
                                 float* __restrict__ S,
                                 float* __restrict__ colsum) {
    __shared__ float lds[16 * 64];
    const int tid   = threadIdx.x;
    const int lane  = tid & 31;
    const int wave  = tid >> 5;
    const int lrow  = lane & 15;
    const int lhalf = lane >> 4;            // 0: K=0,1 / M rows 0..7 ; 1: K=2,3 / rows 8..15
    const int row0  = blockIdx.x * 16;
    const int col0  = wave * 16;

    v8f acc = {0.f, 0.f, 0.f, 0.f, 0.f, 0.f, 0.f, 0.f};
    const float* xrow = x + (size_t)(row0 + lrow) * D_FEAT + 2 * lhalf;
    const float* bp0  = Wa + (size_t)(2 * lhalf) * K_CL + col0 + lrow;
    #pragma unroll 8
    for (int kk = 0; kk < D_FEAT; kk += 4) {
        v2f a = *(const v2f*)(xrow + kk);                  // A[m][k], k=2*lhalf+{0,1}
        v2f b;
        b.x = bp0[(size_t)kk * K_CL];                      // B[k][n] = Wa[k][n]
        b.y = bp0[(size_t)(kk + 1) * K_CL];
        acc = wmma_f32(a, b, acc);
    }
    const float bias = ba[col0 + lrow];
    #pragma unroll
    for (int i = 0; i < 8; ++i)
        lds[(i + 8 * lhalf) * 64 + col0 + lrow] = acc[i] + bias;
    __syncthreads();

    // softmax over 64 cols: 8 threads per row (within one wave), shuffle reduce
    const int row = tid >> 3, h = tid & 7;
    float v[8], m = -3.402823e38f;
    #pragma unroll
    for (int j = 0; j < 8; ++j) { v[j] = lds[row * 64 + h + 8 * j]; m = fmaxf(m, v[j]); }
    for (int off = 4; off >= 1; off >>= 1) m = fmaxf(m, __shfl_xor(m, off, 8));
    float s = 0.f;
    #pragma unroll
    for (int j = 0; j < 8; ++j) { v[j] = __expf(v[j] - m); s += v[j]; }
    for (int off = 4; off >= 1; off >>= 1) s += __shfl_xor(s, off, 8);
    const float inv = 1.f / s;
    #pragma unroll
    for (int j = 0; j < 8; ++j) {
        float val = v[j] * inv;
        lds[row * 64 + h + 8 * j] = val;
        S[(size_t)(row0 + row) * K_CL + h + 8 * j] = val;
    }
    __syncthreads();
    if (tid < K_CL) {                         // one atomic per (block, column)
        float cs = 0.f;
        #pragma unroll
        for (int r = 0; r < 16; ++r) cs += lds[r * 64 + tid];
        atomicAdd(&colsum[tid], cs);
    }
}

// -------- x_t = x @ Wf + bf  (8192x256x256, fp32 WMMA) --------------------
__global__ void k_feat(const float* __restrict__ x, const float* __restrict__ Wf,
                       const float* __restrict__ bf, float* __restrict__ xt) {
    const int tid = threadIdx.x, lane = tid & 31, wave = tid >> 5;
    const int lrow = lane & 15, lhalf = lane >> 4;
    const int row0 = blockIdx.x * 16;
    const int col0 = blockIdx.y * 64 + wave * 16;

    v8f acc = {0.f, 0.f, 0.f, 0.f, 0.f, 0.f, 0.f, 0.f};
    const float* xrow = x + (size_t)(row0 + lrow) * D_FEAT + 2 * lhalf;
    const float* bp0  = Wf + (size_t)(2 * lhalf) * D_FEAT + col0 + lrow;
    #pragma unroll 8
    for (int kk = 0; kk < D_FEAT; kk += 4) {
        v2f a = *(const v2f*)(xrow + kk);
        v2f b;
        b.x = bp0[(size_t)kk * D_FEAT];
        b.y = bp0[(size_t)(kk + 1) * D_FEAT];
        acc = wmma_f32(a, b, acc);
    }
    const float bias = bf[col0 + lrow];
    #pragma unroll
    for (int i = 0; i < 8; ++i)
        xt[(size_t)(row0 + i + 8 * lhalf) * D_FEAT + col0 + lrow] = acc[i] + bias;
}

// -------- T = (B + B^T) @ S without materializing dense A ------------------
// 64 threads per edge (one per cluster column); T and S stay hot in L2.
__global__ void k_edge(const int* __restrict__ ei, const float* __restrict__ ew,
                       const float* __restrict__ S, float* __restrict__ T) {
    const int tid = threadIdx.x;
    const int e = blockIdx.x * 4 + (tid >> 6);
    const int j = tid & 63;
    const int r = ei[e];
    const int c = ei[E_EDGES + e];
    const float w = ew[e];
    atomicAdd(&T[(size_t)r * K_CL + j], w * S[(size_t)c * K_CL + j]);
    atomicAdd(&T[(size_t)c * K_CL + j], w * S[(size_t)r * K_CL + j]);
}

// -------- C[64 x ldb] += S^T(64 x 8192) @ B(8192 x ldb), split-K atomics ---
// One wave per (16x16 tile, K-chunk). Used for pooled_x (B=x_t) and Mraw (B=T).
__global__ void k_pool_gemm(const float* __restrict__ S, const float* __restrict__ B,
                            float* __restrict__ C, int ldb, int chunk) {
    const int lane = threadIdx.x, lrow = lane & 15, lhalf = lane >> 4;
    const int ntiles = ldb >> 4;
    const int m0 = (blockIdx.x / ntiles) * 16;
    const int n0 = (blockIdx.x % ntiles) * 16;
    const int k0 = blockIdx.y * chunk;

    v8f acc = {0.f, 0.f, 0.f, 0.f, 0.f, 0.f, 0.f, 0.f};
    for (int kk = 0; kk < chunk; kk += 4) {
        const int kg = k0 + kk + 2 * lhalf;
        v2f a, b;
        a.x = S[(size_t)kg * K_CL + m0 + lrow];            // A[m][k] = S[k][m]
        a.y = S[(size_t)(kg + 1) * K_CL + m0 + lrow];
        b.x = B[(size_t)kg * ldb + n0 + lrow];
        b.y = B[(size_t)(kg + 1) * ldb + n0 + lrow];
        acc = wmma_f32(a, b, acc);
    }
    #pragma unroll
    for (int i = 0; i < 8; ++i)
        atomicAdd(&C[(size_t)(m0 + i + 8 * lhalf) * ldb + n0 + lrow], acc[i]);
}

// -------- normalize, emit off-diagonal edge weights + indices --------------
__global__ void k_final(const float* __restrict__ Mraw, const float* __restrict__ colsum,
                        float* __restrict__ ei_out, float* __restrict__ ew_out) {
    const int idx = blockIdx.x * blockDim.x + threadIdx.x;
    if (idx >= K_CL * K_CL) return;
    const int i = idx >> 6, j = idx & 63;
    if (i == j) return;
    const int p = i * (K_CL - 1) + (j < i ? j : j - 1);    // row-major off-diagonal order
    ew_out[p] = Mraw[idx] / (colsum[i] + EPS_VAL);
    ei_out[p] = (float)i;
    ei_out[NPAIR + p] = (float)j;
}

extern "C" void kernel_launch(void* const* d_in, const int* in_sizes, int n_in,
                              void* d_out, int out_size, void* d_ws, size_t ws_size,
                              hipStream_t stream) {
    const float* x  = (const float*)d_in[0];
    const int*   ei = (const int*)d_in[1];
    const float* ew = (const float*)d_in[2];
    const float* Wa = (const float*)d_in[3];
    const float* ba = (const float*)d_in[4];
    const float* Wf = (const float*)d_in[5];
    const float* bf = (const float*)d_in[6];

    float* out = (float*)d_out;
    float* px  = out + OFF_PX;
    float* eio = out + OFF_EI;
    float* ewo = out + OFF_EW;
    float* S   = out + OFF_S;

    float* ws   = (float*)d_ws;
    float* xt   = ws + WS_XT;
    float* T    = ws + WS_T;
    float* Mraw = ws + WS_M;
    float* cs   = ws + WS_CS;

    k_zero<<<(N_NODES * K_CL + 255) / 256, 256, 0, stream>>>(T, Mraw, cs, px);
    k_assign_softmax<<<N_NODES / 16, 128, 0, stream>>>(x, Wa, ba, S, cs);
    k_feat<<<dim3(N_NODES / 16, D_FEAT / 64), 128, 0, stream>>>(x, Wf, bf, xt);
    k_edge<<<E_EDGES / 4, 256, 0, stream>>>(ei, ew, S, T);
    // pooled_x = S^T @ x_t : 4x16 tiles, split-K 8 x 1024
    k_pool_gemm<<<dim3(4 * 16, 8), 32, 0, stream>>>(S, xt, px, D_FEAT, 1024);
    // Mraw = S^T @ T : 4x4 tiles, split-K 8 x 1024
    k_pool_gemm<<<dim3(4 * 4, 8), 32, 0, stream>>>(S, T, Mraw, K_CL, 1024);
    k_final<<<(K_CL * K_CL + 255) / 256, 256, 0, stream>>>(Mraw, cs, eio, ewo);
}